// mem_update_64587718197979
// MI455X (gfx1250) — compile-verified
//
#include <hip/hip_runtime.h>

// LIF membrane recurrence:  spikes[t] = (mem[t] > 0.5),
//   mem[t] = 0.25 * mem[t-1] * (1 - spike[t-1]) + x[t],  mem[-1]=spike[-1]=0
//
// x: (T=8, 32,128,32,32) f32.  Pure streaming workload, 256 MB total traffic
// -> HBM-bound (~11 us at 23.3 TB/s).  Strategy: one vec4 lane-column per
// thread, all 8 time-plane B128 loads issued up front (addresses are
// independent of the recurrence), non-temporal hints on both streams since
// every byte is touched exactly once.

typedef float v4f __attribute__((ext_vector_type(4)));

#define T_STEPS 8
#define DECAY   0.25f
#define THRESH  0.5f

__global__ __launch_bounds__(256) void lif_spikes_kernel(
    const v4f* __restrict__ x,   // T_STEPS planes of `stride` vec4s each
    v4f* __restrict__ out,       // same layout
    unsigned int nvec)           // vec4s per time plane (S/4)
{
    const unsigned int gid = blockIdx.x * blockDim.x + threadIdx.x;
    if (gid >= nvec) return;

    // Stage all 8 time planes: 8 independent global_load_b128 (th:NT) in
    // flight per lane before the serial scan starts consuming them.
    v4f xr[T_STEPS];
#pragma unroll
    for (int t = 0; t < T_STEPS; ++t) {
        xr[t] = __builtin_nontemporal_load(&x[(size_t)t * nvec + gid]);
    }

    v4f mem   = {0.0f, 0.0f, 0.0f, 0.0f};
    v4f spike = {0.0f, 0.0f, 0.0f, 0.0f};

#pragma unroll
    for (int t = 0; t < T_STEPS; ++t) {
#pragma unroll
        for (int c = 0; c < 4; ++c) {
            float m  = mem[c] * DECAY * (1.0f - spike[c]) + xr[t][c];
            mem[c]   = m;
            spike[c] = (m > THRESH) ? 1.0f : 0.0f;   // NaN -> 0, matches JAX
        }
        // Streaming output, never re-read by us: global_store_b128 th:NT.
        __builtin_nontemporal_store(spike, &out[(size_t)t * nvec + gid]);
    }
}

extern "C" void kernel_launch(void* const* d_in, const int* in_sizes, int n_in,
                              void* d_out, int out_size, void* d_ws, size_t ws_size,
                              hipStream_t stream) {
    (void)n_in; (void)out_size; (void)d_ws; (void)ws_size;

    const v4f* x  = (const v4f*)d_in[0];
    v4f*      out = (v4f*)d_out;

    // in_sizes[0] = T * S total f32 elements; each thread owns 4 consecutive
    // elements of one spatial column across all T planes.
    const unsigned int total = (unsigned int)in_sizes[0];
    const unsigned int nvec  = total / (T_STEPS * 4u);   // vec4s per plane

    const unsigned int block = 256;                       // 8 wave32s
    const unsigned int grid  = (nvec + block - 1) / block;

    lif_spikes_kernel<<<grid, block, 0, stream>>>(x, out, nvec);
}